// CosineAlignLoss_69827578298843
// MI455X (gfx1250) — compile-verified
//
#include <hip/hip_runtime.h>
#include <hip/hip_bf16.h>

#define HUBER_W 0.4f
#define COS_W   0.3f
#define EPSV    1e-8f

constexpr int BB  = 8;
constexpr int SS  = 2048;
constexpr int TT  = 1024;
constexpr int DD  = 2048;
constexpr int WIN = 7;     // 2*WINDOW + 1

typedef float v2f __attribute__((ext_vector_type(2)));
typedef float v8f __attribute__((ext_vector_type(8)));

// d_ws layout (first 128 bytes zeroed by hipMemsetAsync every launch):
//   f32 ws[0] = huber_sum   f32 ws[1] = cos_sum   f32 ws[2] = count
//   i32 ws[8..15]  = n_s[8]
//   i32 ws[16..23] = n_t[8]

__global__ __launch_bounds__(256)
void mask_sum_kernel(const int* __restrict__ sm, const int* __restrict__ tm,
                     int* __restrict__ wsi) {
  const int which = blockIdx.x;           // 0..7: s_mask rows, 8..15: t_mask rows
  const int* row; int len; int* dst;
  if (which < BB) { row = sm + (size_t)which * SS;        len = SS; dst = wsi + 8  + which; }
  else            { row = tm + (size_t)(which - BB) * TT; len = TT; dst = wsi + 16 + (which - BB); }
  int part = 0;
  for (int i = threadIdx.x; i < len; i += blockDim.x) part += (row[i] == 1);
  #pragma unroll
  for (int off = 16; off > 0; off >>= 1) part += __shfl_xor(part, off, 32);
  if ((threadIdx.x & 31) == 0) atomicAdd(dst, part);
}

__global__ __launch_bounds__(256)
void align_loss_kernel(const float* __restrict__ st, const float* __restrict__ te,
                       const int* __restrict__ tmask,
                       const int* __restrict__ ns, const int* __restrict__ nt,
                       float* __restrict__ acc) {
  const int lane  = threadIdx.x & 31;
  const int token = blockIdx.x * 8 + (threadIdx.x >> 5);   // one wave per (b,t)
  const int b = token >> 10;               // TT == 1024
  const int t = token & (TT - 1);

  const int n_s = ns[b];
  const int n_t = nt[b];
  const int n_t_safe = n_t > 0 ? n_t : 1;
  const int s_pos   = (t * n_s) / n_t_safe;
  const int w_start = s_pos - 3 > 0 ? s_pos - 3 : 0;
  int w_end = s_pos + 4; if (w_end > n_s) w_end = n_s;

  const bool valid = (tmask[(size_t)b * TT + t] == 1) && (n_t > 0) && (n_s > 0)
                     && (w_start < w_end);
  if (!valid) return;                      // wave-uniform exit; EXEC stays all-1s

  // Window rows (wave-uniform values)
  const float* wrow[WIN];
  bool jv[WIN];
  #pragma unroll
  for (int w = 0; w < WIN; ++w) {
    const int j  = w_start + w;
    jv[w] = (j < w_end);
    const int jc = j < SS - 1 ? j : SS - 1;
    wrow[w] = st + ((size_t)b * SS + jc) * (size_t)DD;
  }
  const float* trow = te + ((size_t)b * TT + t) * (size_t)DD;

  // ---- Pass 1: 8x8 Gram of [win_0..win_6, teach] via V_WMMA_F32_16X16X4_F32.
  // 32-bit A layout (16x4): lanes 0-15 hold (M=lane, K={0,1}); lanes 16-31 (M=lane-16, K={2,3}).
  // Feeding the same operand as A and B gives C = M * M^T:
  //   C[w][7] = win_w . teach,  C[r][r] = |row_r|^2.
  // We only ever read C[m][n] with m,n <= 7, and C[m][n] depends ONLY on rows
  // m and n -- so lanes mapping to rows 8..15 may load arbitrary finite data.
  // Point them at the teacher row: every lane then has a uniform 16B/K4 stride,
  // letting the compiler use immediate instruction offsets (no per-load v_add_u64).
  const int r = lane & 15;
  const int khalf = (lane >> 4) << 1;
  const float* p;
  if (r < WIN) {
    const int j  = w_start + r;
    const int jc = j < SS - 1 ? j : SS - 1;
    p = st + ((size_t)b * SS + jc) * (size_t)DD + khalf;
  } else {
    p = trow + khalf;                      // r==7 real teacher; r>=8 harmless filler
  }

  v8f c0 = {0.f}, c1 = {0.f}, c2 = {0.f}, c3 = {0.f};
  for (int k = 0; k < DD; k += 16) {
    v2f a0 = *(const v2f*)(p + 0);
    v2f a1 = *(const v2f*)(p + 4);
    v2f a2 = *(const v2f*)(p + 8);
    v2f a3 = *(const v2f*)(p + 12);
    p += 16;
    c0 = __builtin_amdgcn_wmma_f32_16x16x4_f32(false, a0, false, a0, (short)0, c0, false, false);
    c1 = __builtin_amdgcn_wmma_f32_16x16x4_f32(false, a1, false, a1, (short)0, c1, false, false);
    c2 = __builtin_amdgcn_wmma_f32_16x16x4_f32(false, a2, false, a2, (short)0, c2, false, false);
    c3 = __builtin_amdgcn_wmma_f32_16x16x4_f32(false, a3, false, a3, (short)0, c3, false, false);
  }
  v8f g = (c0 + c1) + (c2 + c3);

  // C(M=m, N=n) lives in VGPR m, lane n (lanes 0-15 block).
  float dotv[WIN], nrm2[WIN];
  #pragma unroll
  for (int w = 0; w < WIN; ++w) {
    dotv[w] = __shfl(g[w], WIN, 32);       // win_w . teach
    nrm2[w] = __shfl(g[w], w,   32);       // |win_w|^2
  }
  const float tn2   = __shfl(g[7], 7, 32); // |teach|^2
  const float tnorm = fmaxf(sqrtf(tn2), EPSV);

  // Softmax over the 7 window slots (replicated in all lanes)
  float logit[WIN], mx = -1e30f;
  #pragma unroll
  for (int w = 0; w < WIN; ++w) {
    const float wn  = fmaxf(sqrtf(nrm2[w]), EPSV);
    const float sim = dotv[w] / (wn * tnorm);
    logit[w] = jv[w] ? sim * 3.0f : -1e30f;
    mx = fmaxf(mx, logit[w]);
  }
  float wt[WIN], se = 0.f;
  #pragma unroll
  for (int w = 0; w < WIN; ++w) { wt[w] = __expf(logit[w] - mx); se += wt[w]; }
  const float inv_se = 1.f / se;
  #pragma unroll
  for (int w = 0; w < WIN; ++w) wt[w] *= inv_se;

  // ---- Pass 2: aligned = sum_w wt[w]*win_w ; huber + cosine terms. Coalesced float4.
  float hsum = 0.f, at = 0.f, aa = 0.f;
  const float4* t4 = (const float4*)trow;
  #pragma unroll 4
  for (int i = 0; i < DD / 128; ++i) {     // 16 iterations, float4 per lane
    const int idx = i * 32 + lane;
    const float4 tv = t4[idx];
    float ax = 0.f, ay = 0.f, az = 0.f, aw = 0.f;
    #pragma unroll
    for (int w = 0; w < WIN; ++w) {
      const float4 wv = ((const float4*)wrow[w])[idx];
      ax = fmaf(wt[w], wv.x, ax);
      ay = fmaf(wt[w], wv.y, ay);
      az = fmaf(wt[w], wv.z, az);
      aw = fmaf(wt[w], wv.w, aw);
    }
    const float d0 = ax - tv.x, d1 = ay - tv.y, d2 = az - tv.z, d3 = aw - tv.w;
    const float b0 = fabsf(d0), b1 = fabsf(d1), b2 = fabsf(d2), b3 = fabsf(d3);
    hsum += (b0 < 1.f ? 0.5f * b0 * b0 : b0 - 0.5f);
    hsum += (b1 < 1.f ? 0.5f * b1 * b1 : b1 - 0.5f);
    hsum += (b2 < 1.f ? 0.5f * b2 * b2 : b2 - 0.5f);
    hsum += (b3 < 1.f ? 0.5f * b3 * b3 : b3 - 0.5f);
    at += ax * tv.x + ay * tv.y + az * tv.z + aw * tv.w;
    aa += ax * ax + ay * ay + az * az + aw * aw;
  }
  #pragma unroll
  for (int off = 16; off > 0; off >>= 1) {
    hsum += __shfl_xor(hsum, off, 32);
    at   += __shfl_xor(at,   off, 32);
    aa   += __shfl_xor(aa,   off, 32);
  }
  if (lane == 0) {
    const float anorm     = fmaxf(sqrtf(aa), EPSV);
    const float cos_tok   = 1.f - at / (anorm * tnorm);
    const float huber_tok = hsum * (1.f / (float)DD);
    atomicAdd(acc + 0, huber_tok);
    atomicAdd(acc + 1, cos_tok);
    atomicAdd(acc + 2, 1.f);
  }
}

__global__ void finalize_kernel(const float* __restrict__ acc, float* __restrict__ out) {
  const float cnt = acc[2];
  const float cs  = fmaxf(cnt, 1.f);
  const bool  has = cnt > 0.f;
  float hm = acc[0] / cs, cm = acc[1] / cs;
  hm = has ? hm : 0.f;
  cm = has ? cm : 0.f;
  out[0] = has ? (HUBER_W * hm + COS_W * cm) : 0.f;
  out[1] = hm;
  out[2] = cm;
  out[3] = cnt;
}

extern "C" void kernel_launch(void* const* d_in, const int* in_sizes, int n_in,
                              void* d_out, int out_size, void* d_ws, size_t ws_size,
                              hipStream_t stream) {
  const float* st = (const float*)d_in[0];   // student_output  [B,S,D]
  const float* te = (const float*)d_in[1];   // teacher_output  [B,T,D]
  // d_in[2] original_student: unused by the reference loss
  const int*   sm = (const int*)d_in[3];     // s_mask [B,S]
  const int*   tm = (const int*)d_in[4];     // t_mask [B,T]
  float* out = (float*)d_out;

  float* wsf = (float*)d_ws;
  int*   wsi = (int*)d_ws;

  hipMemsetAsync(d_ws, 0, 128, stream);      // accumulators + n_s/n_t
  mask_sum_kernel<<<16, 256, 0, stream>>>(sm, tm, wsi);
  align_loss_kernel<<<(BB * TT) / 8, 256, 0, stream>>>(
      st, te, tm, wsi + 8, wsi + 16, wsf);
  finalize_kernel<<<1, 1, 0, stream>>>(wsf, out);
}